// WhiteboxGATLayer_85177791414408
// MI455X (gfx1250) — compile-verified
//
#include <hip/hip_runtime.h>
#include <math.h>

#define N      4096
#define DIM    256
#define OUTD   128
#define HEADS  4
#define SUB    32
#define ETA    0.5f
#define NEGBIG 1.0e9f
#define RSQRTS 0.17677669529663687f  /* 1/sqrt(32) */

typedef float v2f __attribute__((ext_vector_type(2)));
typedef float v8f __attribute__((ext_vector_type(8)));

__device__ __forceinline__ v8f wmma4(v2f a, v2f b, v8f c) {
  // D(16x16,f32) = A(16x4,f32) x B(4x16,f32) + C
  return __builtin_amdgcn_wmma_f32_16x16x4_f32(false, a, false, b, (short)0, c,
                                               false, false);
}

// CDNA5 async global->LDS DMA (ASYNCcnt). LDS operand = addr[31:0] of the
// flat address.
__device__ __forceinline__ void ldsAsyncB128(void* lds, const void* gaddr) {
  unsigned off = (unsigned)(uintptr_t)lds;
  asm volatile("global_load_async_to_lds_b128 %0, %1, off"
               :: "v"(off), "v"(gaddr) : "memory");
}
__device__ __forceinline__ void waitAsync0() {
  asm volatile("s_wait_asynccnt 0x0" ::: "memory");
}

__device__ __forceinline__ float hmax16(float v) {
  v = fmaxf(v, __shfl_xor(v, 1, 16));
  v = fmaxf(v, __shfl_xor(v, 2, 16));
  v = fmaxf(v, __shfl_xor(v, 4, 16));
  v = fmaxf(v, __shfl_xor(v, 8, 16));
  return v;
}
__device__ __forceinline__ float hsum16(float v) {
  v += __shfl_xor(v, 1, 16);
  v += __shfl_xor(v, 2, 16);
  v += __shfl_xor(v, 4, 16);
  v += __shfl_xor(v, 8, 16);
  return v;
}

// ---------------- prep kernels ----------------
__global__ __launch_bounds__(256) void k_pack_adj(const float* __restrict__ adj,
                                                  unsigned* __restrict__ bits) {
  int w = blockIdx.x * blockDim.x + threadIdx.x;
  if (w >= N * (N / 32)) return;
  int row = w >> 7;               // N/32 = 128 words per row
  int c0  = (w & 127) * 32;
  const float* p = adj + (size_t)row * N + c0;
  unsigned b = 0u;
#pragma unroll
  for (int i = 0; i < 32; ++i) b |= (p[i] != 0.0f ? 1u : 0u) << i;
  bits[w] = b;
}

__global__ __launch_bounds__(256) void k_transpose_u(const float* __restrict__ U,
                                                     float* __restrict__ Ut) {
  int t = blockIdx.x * blockDim.x + threadIdx.x;   // 128*256
  if (t >= OUTD * DIM) return;
  int p = t >> 8, d = t & 255;                     // p = h*32+s
  Ut[p * DIM + d] = U[((p >> 5) * DIM + d) * SUB + (p & 31)];
}

__global__ __launch_bounds__(256) void k_transpose_p(const float* __restrict__ Pw,
                                                     float* __restrict__ Pt) {
  int t = blockIdx.x * blockDim.x + threadIdx.x;   // 256*128
  if (t >= DIM * OUTD) return;
  int d = t >> 7, o = t & 127;
  Pt[d * OUTD + o] = Pw[o * DIM + d];
}

// ---------------- kernel 1: Z = H @ Ucat  -> Zt[h][n][s] ----------------
__global__ __launch_bounds__(128) void k_zproj(const float* __restrict__ H,
                                               const float* __restrict__ U,
                                               float* __restrict__ Zt) {
  __shared__ float HL[16 * 260];   // 1040B row pitch (16B aligned)
  const int I = blockIdx.x * 16;
  const int tid = threadIdx.x;
#pragma unroll
  for (int i = 0; i < 8; ++i) {
    int idx = (tid * 8 + i) * 4;   // 16*256 floats, 4 per async b128
    int r = idx >> 8, c = idx & 255;
    ldsAsyncB128(&HL[r * 260 + c], &H[(size_t)(I + r) * DIM + c]);
  }
  waitAsync0();
  __syncthreads();
  const int wave = tid >> 5, lane = tid & 31;
  const int lrow = lane & 15, hi = lane >> 4;
#pragma unroll
  for (int t2 = 0; t2 < 2; ++t2) {
    int tile = wave * 2 + t2;            // 0..7 : (head, s-half)
    int h = tile >> 1;
    int s = (tile & 1) * 16 + lrow;
    v8f c8 = {};
#pragma unroll 4
    for (int cc = 0; cc < 64; ++cc) {
      int k = cc * 4 + hi * 2;
      v2f a = *(const v2f*)&HL[lrow * 260 + k];
      v2f b;
      b[0] = U[(size_t)(h * DIM + k) * SUB + s];
      b[1] = U[(size_t)(h * DIM + k + 1) * SUB + s];
      c8 = wmma4(a, b, c8);
    }
#pragma unroll
    for (int r = 0; r < 8; ++r) {
      int row = I + r + 8 * hi;
      Zt[((size_t)h * N + row) * SUB + (tile & 1) * 16 + lrow] = c8[r];
    }
  }
}

// ---------------- kernel 2: fused masked-softmax attention ----------------
#define ZP 36   // LDS pitch (144B rows: 16B-aligned b128 targets, bank-safe)
#define PP 18   // LDS pitch for 16-wide P tile
__device__ __forceinline__ void stageZ(float* buf, const float* Zh, int J, int tid) {
#pragma unroll
  for (int i = 0; i < 2; ++i) {
    int idx = (tid * 2 + i) * 4;      // 32*32 floats, 4 per async b128
    int r = idx >> 5, c = idx & 31;
    ldsAsyncB128(&buf[r * ZP + c], &Zh[(size_t)(J + r) * SUB + c]);
  }
}

__global__ __launch_bounds__(128) void k_attn(const float* __restrict__ Zt,
                                              const unsigned* __restrict__ bits,
                                              float* __restrict__ Agg) {
  __shared__ float ZL[2][32 * ZP];
  __shared__ float PL[4][16 * PP];
  const int h   = blockIdx.x >> 6;          // 4 heads x 64 groups
  const int blk = blockIdx.x & 63;          // 64 rows per group
  const int tid = threadIdx.x, wave = tid >> 5, lane = tid & 31;
  const int lrow = lane & 15, hi = lane >> 4;
  const int I = blk * 64 + wave * 16;
  const float* Zh = Zt + (size_t)h * N * SUB;
  float* myP = &PL[wave][0];

  // Q in fp32 A-layout: 16 regs/lane
  float qa[16];
#pragma unroll
  for (int c = 0; c < 8; ++c) {
    v2f t = *(const v2f*)&Zh[(size_t)(I + lrow) * SUB + c * 4 + hi * 2];
    qa[2 * c] = t[0]; qa[2 * c + 1] = t[1];
  }

  float m[8], l[8];
  v8f acc0 = {}, acc1 = {};
#pragma unroll
  for (int r = 0; r < 8; ++r) { m[r] = -3.0e38f; l[r] = 0.0f; }

  stageZ(ZL[0], Zh, 0, tid);               // prime pipeline
  int cur = 0;
  for (int J = 0; J < N; J += 32) {
    waitAsync0();                          // my async fills of ZL[cur] done
    __syncthreads();                       // everyone's done; prev reads done
    if (J + 32 < N) stageZ(ZL[cur ^ 1], Zh, J + 32, tid);  // prefetch next
    const float* Zc = ZL[cur];

    unsigned w[8];
#pragma unroll
    for (int r = 0; r < 8; ++r)
      w[r] = bits[(size_t)(I + r + 8 * hi) * (N / 32) + (J >> 5)];

#pragma unroll
    for (int t = 0; t < 2; ++t) {
      // scores: Sc = Q(16x32) @ Kt(32x16) / sqrt(S)
      v8f sc = {};
#pragma unroll
      for (int c = 0; c < 8; ++c) {
        v2f a; a[0] = qa[2 * c]; a[1] = qa[2 * c + 1];
        int k = c * 4 + hi * 2;
        v2f b = *(const v2f*)&Zc[(t * 16 + lrow) * ZP + k];
        sc = wmma4(a, b, sc);
      }
      // mask + online softmax (per row r+8*hi)
      float corr[8];
#pragma unroll
      for (int r = 0; r < 8; ++r) {
        float s = sc[r] * RSQRTS;
        unsigned bit = (w[r] >> (t * 16 + lrow)) & 1u;
        s = bit ? s : s - NEGBIG;
        float nm = fmaxf(m[r], hmax16(s));
        float cr = __expf(m[r] - nm);
        float pv = __expf(s - nm);
        l[r] = l[r] * cr + hsum16(pv);
        m[r] = nm;
        corr[r] = cr;
        myP[(r + 8 * hi) * PP + lrow] = pv;   // C-layout -> LDS
      }
#pragma unroll
      for (int r = 0; r < 8; ++r) { acc0[r] *= corr[r]; acc1[r] *= corr[r]; }
      // acc(16x32) += P(16x16) @ V(16x32)
#pragma unroll
      for (int c = 0; c < 4; ++c) {
        int j = c * 4 + hi * 2;
        v2f a = *(const v2f*)&myP[lrow * PP + j];  // A-layout from LDS
        v2f b0, b1;
        b0[0] = Zc[(t * 16 + j) * ZP + lrow];
        b0[1] = Zc[(t * 16 + j + 1) * ZP + lrow];
        acc0 = wmma4(a, b0, acc0);
        b1[0] = Zc[(t * 16 + j) * ZP + 16 + lrow];
        b1[1] = Zc[(t * 16 + j + 1) * ZP + 16 + lrow];
        acc1 = wmma4(a, b1, acc1);
      }
    }
    cur ^= 1;
  }
#pragma unroll
  for (int r = 0; r < 8; ++r) {
    float inv = 1.0f / l[r];
    int row = I + r + 8 * hi;
    Agg[(size_t)row * OUTD + h * SUB + lrow]      = acc0[r] * inv;
    Agg[(size_t)row * OUTD + h * SUB + 16 + lrow] = acc1[r] * inv;
  }
}

// ------- kernel 3: H_out = softthresh((H + eta*Agg@Ut) @ proj^T) -------
__global__ __launch_bounds__(128) void k_proj(const float* __restrict__ H,
                                              const float* __restrict__ Agg,
                                              const float* __restrict__ Ut,
                                              const float* __restrict__ Pt,
                                              const float* __restrict__ thr,
                                              float* __restrict__ out) {
  __shared__ float AL[16 * 132];   // 528B rows (16B aligned)
  __shared__ float ML[16 * 260];
  const int I = blockIdx.x * 16;
  const int tid = threadIdx.x, wave = tid >> 5, lane = tid & 31;
  const int lrow = lane & 15, hi = lane >> 4;
#pragma unroll
  for (int i = 0; i < 4; ++i) {
    int idx = (tid * 4 + i) * 4;   // 16*128 floats, 4 per async b128
    int r = idx >> 7, c = idx & 127;
    ldsAsyncB128(&AL[r * 132 + c], &Agg[(size_t)(I + r) * OUTD + c]);
  }
  waitAsync0();
  __syncthreads();
  // GEMM1: M1 = H + eta * Agg(16x128) @ Ut(128x256)
#pragma unroll
  for (int tt = 0; tt < 4; ++tt) {
    int dt = wave + tt * 4;          // d-tile 0..15
    v8f c8 = {};
#pragma unroll 4
    for (int c = 0; c < 32; ++c) {
      int k = c * 4 + hi * 2;
      v2f a = *(const v2f*)&AL[lrow * 132 + k];
      v2f b;
      b[0] = Ut[(size_t)k * DIM + dt * 16 + lrow];
      b[1] = Ut[(size_t)(k + 1) * DIM + dt * 16 + lrow];
      c8 = wmma4(a, b, c8);
    }
#pragma unroll
    for (int r = 0; r < 8; ++r) {
      int row = r + 8 * hi;
      float hv = H[(size_t)(I + row) * DIM + dt * 16 + lrow];
      ML[row * 260 + dt * 16 + lrow] = hv + ETA * c8[r];
    }
  }
  __syncthreads();
  // GEMM2: out = softthresh(M1(16x256) @ Pt(256x128))
#pragma unroll
  for (int tt = 0; tt < 2; ++tt) {
    int ot = wave * 2 + tt;          // out-tile 0..7
    v8f c8 = {};
#pragma unroll 4
    for (int c = 0; c < 64; ++c) {
      int k = c * 4 + hi * 2;
      v2f a = *(const v2f*)&ML[lrow * 260 + k];
      v2f b;
      b[0] = Pt[(size_t)k * OUTD + ot * 16 + lrow];
      b[1] = Pt[(size_t)(k + 1) * OUTD + ot * 16 + lrow];
      c8 = wmma4(a, b, c8);
    }
    float th = thr[ot * 16 + lrow];
#pragma unroll
    for (int r = 0; r < 8; ++r) {
      float v = c8[r];
      float a = fabsf(v) - th;
      out[(size_t)(I + r + 8 * hi) * OUTD + ot * 16 + lrow] =
          (a > 0.0f) ? copysignf(a, v) : 0.0f;
    }
  }
}

// ---------------- kernel 4: orthogonality loss ----------------
__global__ __launch_bounds__(256) void k_orth(const float* __restrict__ Ut,
                                              float* __restrict__ out) {
  __shared__ float red[256];
  const int tid = threadIdx.x, wave = tid >> 5, lane = tid & 31;
  const int lrow = lane & 15, hi = lane >> 4;
  float local = 0.0f;
  for (int t = 0; t < 8; ++t) {
    int idx = wave * 8 + t;
    int ti = idx >> 3, tj = idx & 7;        // 16-row tiles of the 128x128 Gram
    if ((ti >> 1) > (tj >> 1)) continue;    // keep head-blocks k<=l (triu)
    v8f c8 = {};
#pragma unroll 4
    for (int c = 0; c < 64; ++c) {
      int k = c * 4 + hi * 2;
      v2f a = *(const v2f*)&Ut[(size_t)(ti * 16 + lrow) * DIM + k];
      v2f b = *(const v2f*)&Ut[(size_t)(tj * 16 + lrow) * DIM + k];
      c8 = wmma4(a, b, c8);
    }
#pragma unroll
    for (int r = 0; r < 8; ++r) {
      int p = ti * 16 + r + 8 * hi;
      int q = tj * 16 + lrow;
      float v = c8[r] - ((p == q) ? 1.0f : 0.0f);
      local += v * v;
    }
  }
  red[tid] = local;
  __syncthreads();
  for (int s = 128; s > 0; s >>= 1) {
    if (tid < s) red[tid] += red[tid + s];
    __syncthreads();
  }
  if (tid == 0) out[(size_t)N * OUTD] = red[0];
}

// ---------------- launch ----------------
extern "C" void kernel_launch(void* const* d_in, const int* in_sizes, int n_in,
                              void* d_out, int out_size, void* d_ws, size_t ws_size,
                              hipStream_t stream) {
  (void)in_sizes; (void)n_in; (void)out_size; (void)ws_size;
  const float* H   = (const float*)d_in[0];
  const float* adj = (const float*)d_in[1];
  const float* U   = (const float*)d_in[2];
  const float* thr = (const float*)d_in[3];
  const float* pw  = (const float*)d_in[4];
  float* out = (float*)d_out;

  char* ws = (char*)d_ws;
  float*    Zt   = (float*)(ws);                                   // 2 MB
  float*    Agg  = (float*)(ws + (size_t)(2 << 20));               // 2 MB
  unsigned* bits = (unsigned*)(ws + (size_t)(4 << 20));            // 2 MB
  float*    Ut   = (float*)(ws + (size_t)(6 << 20));               // 128 KB
  float*    Pt   = (float*)(ws + (size_t)(6 << 20) + (128 << 10)); // 128 KB

  k_pack_adj   <<<(N * (N / 32) + 255) / 256, 256, 0, stream>>>(adj, bits);
  k_transpose_u<<<(OUTD * DIM + 255) / 256,   256, 0, stream>>>(U, Ut);
  k_transpose_p<<<(DIM * OUTD + 255) / 256,   256, 0, stream>>>(pw, Pt);
  k_zproj      <<<N / 16, 128, 0, stream>>>(H, U, Zt);
  k_attn       <<<HEADS * (N / 64), 128, 0, stream>>>(Zt, bits, Agg);
  k_proj       <<<N / 16, 128, 0, stream>>>(H, Agg, Ut, Pt, thr, out);
  k_orth       <<<1, 256, 0, stream>>>(Ut, out);
}